// GNN_82171314307289
// MI455X (gfx1250) — compile-verified
//
#include <hip/hip_runtime.h>
#include <math.h>

typedef __attribute__((ext_vector_type(2))) float v2f;
typedef __attribute__((ext_vector_type(8))) float v8f;

// ---------------------------------------------------------------------------
// WMMA helper: exact fp32 16x16x4 matrix FMA on the CDNA5 matrix pipe.
// A: 16x4 (lane half selects K-pair, vgpr selects K parity), B: 4x16, C/D 16x16.
// ---------------------------------------------------------------------------
__device__ __forceinline__ v8f wmma4(v2f a, v2f b, v8f c) {
  return __builtin_amdgcn_wmma_f32_16x16x4_f32(false, a, false, b, (short)0, c,
                                               false, false);
}

__device__ __forceinline__ void atomicMaxFloat(float* addr, float val) {
  if (val >= 0.f)
    atomicMax((int*)addr, __float_as_int(val));
  else
    atomicMin((unsigned int*)addr, __float_as_uint(val));
}

// ---------------------------------------------------------------------------
// utility
// ---------------------------------------------------------------------------
__global__ void k_fill(float* p, float v, long n) {
  long i = (long)blockIdx.x * blockDim.x + threadIdx.x;
  if (i < n) p[i] = v;
}

// in-degree counts (float)
__global__ void k_count(const int* __restrict__ dst, float* cnt, int E) {
  int i = blockIdx.x * blockDim.x + threadIdx.x;
  if (i < E) atomicAdd(&cnt[dst[i]], 1.0f);
}

// ---------------------------------------------------------------------------
// SAGE layer 1 (K=3, scalar path is optimal)
// ---------------------------------------------------------------------------
__global__ void k_s1agg(const float* __restrict__ pos, const int* __restrict__ src,
                        const int* __restrict__ dst, float* agg, int E) {
  int i = blockIdx.x * blockDim.x + threadIdx.x;
  if (i >= E * 3) return;
  int e = i / 3, c = i - e * 3;
  atomicAdd(&agg[(size_t)dst[e] * 3 + c], pos[(size_t)src[e] * 3 + c]);
}

__global__ void k_sage1(const float* __restrict__ pos, const float* __restrict__ agg,
                        const float* __restrict__ cnt, const float* __restrict__ wl,
                        const float* __restrict__ wr, const float* __restrict__ bias,
                        float* __restrict__ x1, int N) {
  int i = blockIdx.x * blockDim.x + threadIdx.x;
  if (i >= N * 128) return;
  int n = i >> 7, j = i & 127;
  float ic = 1.f / fmaxf(cnt[n], 1.f);
  float s = bias[j];
#pragma unroll
  for (int k = 0; k < 3; ++k)
    s += agg[(size_t)n * 3 + k] * ic * wl[j * 3 + k] + pos[(size_t)n * 3 + k] * wr[j * 3 + k];
  x1[i] = fmaxf(s, 0.f);
}

// edge scatter of 128-channel features (mean aggregation numerator)
__global__ void k_scatter(const float* __restrict__ x, const int* __restrict__ src,
                          const int* __restrict__ dst, float* agg) {
  int e = blockIdx.x, c = threadIdx.x;
  atomicAdd(&agg[(size_t)dst[e] * 128 + c], x[(size_t)src[e] * 128 + c]);
}

__global__ void k_div(float* agg, const float* __restrict__ cnt, int N) {
  int i = blockIdx.x * blockDim.x + threadIdx.x;
  if (i >= N * 128) return;
  agg[i] *= 1.f / fmaxf(cnt[i >> 7], 1.f);
}

// ---------------------------------------------------------------------------
// SAGE layer 2: out = relu(agg @ wl^T + x1 @ wr^T + b), fused dual WMMA GEMM.
// Block = 256 threads = 8 waves; wave w -> output cols [16w,16w+16); 16 rows/blk.
// ---------------------------------------------------------------------------
__global__ void k_sage2(const float* __restrict__ agg, const float* __restrict__ x1,
                        const float* __restrict__ wl, const float* __restrict__ wr,
                        const float* __restrict__ bias, float* __restrict__ out, int N) {
  const int lane = threadIdx.x & 31;
  const int wv = threadIdx.x >> 5;
  const int row0 = blockIdx.x * 16;
  const int col0 = wv * 16;
  const int m = lane & 15;
  const int hf = lane >> 4;
  const int ar = min(row0 + m, N - 1);
  const float* A0 = agg + (size_t)ar * 128 + 2 * hf;
  const float* A1 = x1 + (size_t)ar * 128 + 2 * hf;
  const float* B0 = wl + (size_t)(col0 + m) * 128 + 2 * hf;
  const float* B1 = wr + (size_t)(col0 + m) * 128 + 2 * hf;
  v8f c = {};
  for (int k0 = 0; k0 < 128; k0 += 4) {
    v2f a, b;
    a.x = A0[k0]; a.y = A0[k0 + 1];
    b.x = B0[k0]; b.y = B0[k0 + 1];
    c = wmma4(a, b, c);
  }
  for (int k0 = 0; k0 < 128; k0 += 4) {
    v2f a, b;
    a.x = A1[k0]; a.y = A1[k0 + 1];
    b.x = B1[k0]; b.y = B1[k0 + 1];
    c = wmma4(a, b, c);
  }
  float bv = bias[col0 + m];
#pragma unroll
  for (int r = 0; r < 8; ++r) {
    int row = row0 + r + 8 * hf;
    if (row < N) out[(size_t)row * 128 + col0 + m] = fmaxf(c[r] + bv, 0.f);
  }
}

// ---------------------------------------------------------------------------
// GAT feature transform: xw = concat(x1,x2) @ gw^T  (K=256, 256 out channels)
// gridDim.y selects which 128 output columns this block covers.
// ---------------------------------------------------------------------------
__global__ void k_gatw(const float* __restrict__ x1, const float* __restrict__ x2,
                       const float* __restrict__ gw, float* __restrict__ xw, int N) {
  const int lane = threadIdx.x & 31;
  const int wv = threadIdx.x >> 5;
  const int row0 = blockIdx.x * 16;
  const int col0 = blockIdx.y * 128 + wv * 16;
  const int m = lane & 15;
  const int hf = lane >> 4;
  const int ar = min(row0 + m, N - 1);
  const float* A0 = x1 + (size_t)ar * 128 + 2 * hf;
  const float* A1 = x2 + (size_t)ar * 128 + 2 * hf;
  const float* B = gw + (size_t)(col0 + m) * 256 + 2 * hf;
  v8f c = {};
  for (int k0 = 0; k0 < 128; k0 += 4) {
    v2f a, b;
    a.x = A0[k0]; a.y = A0[k0 + 1];
    b.x = B[k0];  b.y = B[k0 + 1];
    c = wmma4(a, b, c);
  }
  for (int k0 = 0; k0 < 128; k0 += 4) {
    v2f a, b;
    a.x = A1[k0];      a.y = A1[k0 + 1];
    b.x = B[128 + k0]; b.y = B[128 + k0 + 1];
    c = wmma4(a, b, c);
  }
#pragma unroll
  for (int r = 0; r < 8; ++r) {
    int row = row0 + r + 8 * hf;
    if (row < N) xw[(size_t)row * 256 + col0 + m] = c[r];
  }
}

// per-(node,head) attention coefficients: wave-per-pair reduction
__global__ void k_attn(const float* __restrict__ xw, const float* __restrict__ gas,
                       const float* __restrict__ gad, float* asrc, float* adst, int N) {
  int gwave = (blockIdx.x * blockDim.x + threadIdx.x) >> 5;
  int lane = threadIdx.x & 31;
  if (gwave >= 2 * N) return;
  int n = gwave >> 1, h = gwave & 1;
  const float* xr = xw + (size_t)n * 256 + h * 128;
  const float* ws = gas + h * 128;
  const float* wd = gad + h * 128;
  float ss = 0.f, sd = 0.f;
  for (int i = lane; i < 128; i += 32) {
    float v = xr[i];
    ss += v * ws[i];
    sd += v * wd[i];
  }
  for (int off = 16; off; off >>= 1) {
    ss += __shfl_down(ss, off);
    sd += __shfl_down(sd, off);
  }
  if (lane == 0) {
    asrc[n * 2 + h] = ss;
    adst[n * 2 + h] = sd;
  }
}

__device__ __forceinline__ void edge_sd(const int* src, const int* dst, int E,
                                        long e, int& s, int& d) {
  if (e < E) { s = src[e]; d = dst[e]; }
  else { s = (int)(e - E); d = s; }
}

__global__ void k_edge_max(const int* __restrict__ src, const int* __restrict__ dst,
                           int E, int N, const float* __restrict__ asrc,
                           const float* __restrict__ adst, float* emax) {
  long i = (long)blockIdx.x * blockDim.x + threadIdx.x;
  if (i >= 2L * (E + N)) return;
  int h = (int)(i & 1); long e = i >> 1;
  int s, d; edge_sd(src, dst, E, e, s, d);
  float v = asrc[s * 2 + h] + adst[d * 2 + h];
  v = v > 0.f ? v : 0.2f * v;
  atomicMaxFloat(&emax[d * 2 + h], v);
}

__global__ void k_edge_sum(const int* __restrict__ src, const int* __restrict__ dst,
                           int E, int N, const float* __restrict__ asrc,
                           const float* __restrict__ adst,
                           const float* __restrict__ emax, float* denom) {
  long i = (long)blockIdx.x * blockDim.x + threadIdx.x;
  if (i >= 2L * (E + N)) return;
  int h = (int)(i & 1); long e = i >> 1;
  int s, d; edge_sd(src, dst, E, e, s, d);
  float v = asrc[s * 2 + h] + adst[d * 2 + h];
  v = v > 0.f ? v : 0.2f * v;
  atomicAdd(&denom[d * 2 + h], expf(v - emax[d * 2 + h]));
}

// grid = E+N blocks, 256 threads = (head, channel)
__global__ void k_edge_msg(const int* __restrict__ src, const int* __restrict__ dst,
                           int E, int N, const float* __restrict__ asrc,
                           const float* __restrict__ adst,
                           const float* __restrict__ emax,
                           const float* __restrict__ denom,
                           const float* __restrict__ xw, float* gacc) {
  long e = blockIdx.x;
  int t = threadIdx.x;
  int h = t >> 7;
  int s, d; edge_sd(src, dst, E, e, s, d);
  float v = asrc[s * 2 + h] + adst[d * 2 + h];
  v = v > 0.f ? v : 0.2f * v;
  float alpha = expf(v - emax[d * 2 + h]) / denom[d * 2 + h];
  atomicAdd(&gacc[(size_t)d * 256 + t], alpha * xw[(size_t)s * 256 + t]);
}

// mean over incoming edges (cnt+1 incl. self loop), then mean over heads + bias
__global__ void k_gat_final(const float* __restrict__ gacc, const float* __restrict__ cnt,
                            const float* __restrict__ gb, float* x3g, int N) {
  int i = blockIdx.x * blockDim.x + threadIdx.x;
  if (i >= N * 128) return;
  int n = i >> 7, c = i & 127;
  float invc = 1.f / (cnt[n] + 1.f);
  x3g[i] = 0.5f * (gacc[(size_t)n * 256 + c] + gacc[(size_t)n * 256 + 128 + c]) * invc + gb[c];
}

// ---------------------------------------------------------------------------
// global mean pool + decoder
// ---------------------------------------------------------------------------
__global__ void k_pool(const float* __restrict__ x3g, const int* __restrict__ batch,
                       float* zsum, float* bcnt, int N) {
  int i = blockIdx.x * blockDim.x + threadIdx.x;
  if (i >= N * 128) return;
  int n = i >> 7, c = i & 127;
  int b = batch[n];
  atomicAdd(&zsum[b * 128 + c], x3g[i]);
  if (c == 0) atomicAdd(&bcnt[b], 1.f);
}

__global__ void k_pool_div(const float* __restrict__ zsum, const float* __restrict__ bcnt,
                           float* z0, int B) {
  int i = blockIdx.x * blockDim.x + threadIdx.x;
  if (i >= B * 128) return;
  z0[i] = zsum[i] / fmaxf(bcnt[i >> 7], 1.f);
}

__global__ void k_linear(const float* __restrict__ x, const float* __restrict__ w,
                         const float* __restrict__ bias, float* y, int M, int K, int Nout) {
  int i = blockIdx.x * blockDim.x + threadIdx.x;
  if (i >= M * Nout) return;
  int r = i / Nout, c = i - r * Nout;
  float s = bias[c];
  for (int k = 0; k < K; ++k) s += x[r * K + k] * w[c * K + k];
  y[i] = s;
}

__global__ void k_bn(float* x, const float* __restrict__ g, const float* __restrict__ b,
                     int M, int C, int relu) {
  int c = blockIdx.x * blockDim.x + threadIdx.x;
  if (c >= C) return;
  float m = 0.f, m2 = 0.f;
  for (int r = 0; r < M; ++r) {
    float v = x[r * C + c];
    m += v; m2 += v * v;
  }
  m /= (float)M; m2 /= (float)M;
  float inv = 1.f / sqrtf(m2 - m * m + 1e-5f);
  for (int r = 0; r < M; ++r) {
    float v = g[c] * (x[r * C + c] - m) * inv + b[c];
    if (relu) v = fmaxf(v, 0.f);
    x[r * C + c] = v;
  }
}

__global__ void k_argmax(const float* __restrict__ z, float* out, int M, int C) {
  int r = blockIdx.x * blockDim.x + threadIdx.x;
  if (r >= M) return;
  float best = z[r * C]; int bi = 0;
  for (int c = 1; c < C; ++c)
    if (z[r * C + c] > best) { best = z[r * C + c]; bi = c; }
  out[r] = (float)bi;
}

// ---------------------------------------------------------------------------
extern "C" void kernel_launch(void* const* d_in, const int* in_sizes, int n_in,
                              void* d_out, int out_size, void* d_ws, size_t ws_size,
                              hipStream_t stream) {
  (void)n_in; (void)out_size; (void)ws_size;
  const float* pos   = (const float*)d_in[0];
  const int*   ei    = (const int*)d_in[1];
  const int*   batch = (const int*)d_in[2];
  const float* s1wl  = (const float*)d_in[3];
  const float* s1wr  = (const float*)d_in[4];
  const float* s1b   = (const float*)d_in[5];
  const float* s2wl  = (const float*)d_in[6];
  const float* s2wr  = (const float*)d_in[7];
  const float* s2b   = (const float*)d_in[8];
  const float* gw    = (const float*)d_in[9];
  const float* gas   = (const float*)d_in[10];
  const float* gad   = (const float*)d_in[11];
  const float* gb    = (const float*)d_in[12];
  const float* d1w   = (const float*)d_in[13];
  const float* d1b   = (const float*)d_in[14];
  const float* bn1g  = (const float*)d_in[15];
  const float* bn1b  = (const float*)d_in[16];
  const float* d2w   = (const float*)d_in[17];
  const float* d2b   = (const float*)d_in[18];
  const float* bn2g  = (const float*)d_in[19];
  const float* bn2b  = (const float*)d_in[20];
  const float* d3w   = (const float*)d_in[21];
  const float* d3b   = (const float*)d_in[22];
  const float* bn3g  = (const float*)d_in[23];
  const float* bn3b  = (const float*)d_in[24];

  const int N = in_sizes[0] / 3;
  const int E = in_sizes[1] / 2;
  const int Bz = 64, CLS = 40;
  const int* src = ei;
  const int* dst = ei + E;

  // workspace layout (floats); gacc aliases x1+x2 (dead after xw),
  // x3g aliases agg (dead after x2). Peak ~260 MB.
  float* wsf  = (float*)d_ws;
  const size_t Ns = (size_t)N;
  float* x1   = wsf;                 // 128N
  float* x2   = wsf + 128 * Ns;      // 128N
  float* gacc = x1;                  // 256N (reuse)
  float* xw   = wsf + 256 * Ns;      // 256N
  float* agg  = wsf + 512 * Ns;      // 128N (agg1 uses first 3N; later x3g)
  float* cnt  = wsf + 640 * Ns;      // N
  float* asrc = wsf + 641 * Ns;      // 2N
  float* adst = wsf + 643 * Ns;      // 2N
  float* emax = wsf + 645 * Ns;      // 2N
  float* denm = wsf + 647 * Ns;      // 2N
  float* zsum = wsf + 649 * Ns;
  float* bcnt = zsum + Bz * 128;
  float* z0   = bcnt + Bz;
  float* t1   = z0 + Bz * 128;
  float* t2   = t1 + Bz * 256;
  float* zout = (float*)d_out;

  auto nb = [](long n, int t) { return dim3((unsigned)((n + t - 1) / t)); };

  // degree counts + SAGE1
  k_fill<<<nb(N, 256), 256, 0, stream>>>(cnt, 0.f, N);
  k_fill<<<nb(3L * N, 256), 256, 0, stream>>>(agg, 0.f, 3L * N);
  k_count<<<nb(E, 256), 256, 0, stream>>>(dst, cnt, E);
  k_s1agg<<<nb(3L * E, 256), 256, 0, stream>>>(pos, src, dst, agg, E);
  k_sage1<<<nb(128L * N, 256), 256, 0, stream>>>(pos, agg, cnt, s1wl, s1wr, s1b, x1, N);

  // SAGE2 (WMMA)
  k_fill<<<nb(128L * N, 256), 256, 0, stream>>>(agg, 0.f, 128L * N);
  k_scatter<<<dim3((unsigned)E), 128, 0, stream>>>(x1, src, dst, agg);
  k_div<<<nb(128L * N, 256), 256, 0, stream>>>(agg, cnt, N);
  k_sage2<<<dim3((unsigned)((N + 15) / 16)), 256, 0, stream>>>(agg, x1, s2wl, s2wr, s2b, x2, N);

  // GAT transform (WMMA) + edge softmax + mean aggregation
  k_gatw<<<dim3((unsigned)((N + 15) / 16), 2), 256, 0, stream>>>(x1, x2, gw, xw, N);
  k_attn<<<nb(2L * N * 32, 256), 256, 0, stream>>>(xw, gas, gad, asrc, adst, N);
  k_fill<<<nb(2L * N, 256), 256, 0, stream>>>(emax, -3.0e38f, 2L * N);
  k_fill<<<nb(2L * N, 256), 256, 0, stream>>>(denm, 0.f, 2L * N);
  k_edge_max<<<nb(2L * (E + N), 256), 256, 0, stream>>>(src, dst, E, N, asrc, adst, emax);
  k_edge_sum<<<nb(2L * (E + N), 256), 256, 0, stream>>>(src, dst, E, N, asrc, adst, emax, denm);
  k_fill<<<nb(256L * N, 256), 256, 0, stream>>>(gacc, 0.f, 256L * N);
  k_edge_msg<<<dim3((unsigned)(E + N)), 256, 0, stream>>>(src, dst, E, N, asrc, adst, emax, denm, xw, gacc);
  k_gat_final<<<nb(128L * N, 256), 256, 0, stream>>>(gacc, cnt, gb, agg, N);

  // pool + decoder
  k_fill<<<nb(Bz * 128, 256), 256, 0, stream>>>(zsum, 0.f, Bz * 128);
  k_fill<<<dim3(1), 64, 0, stream>>>(bcnt, 0.f, Bz);
  k_pool<<<nb(128L * N, 256), 256, 0, stream>>>(agg, batch, zsum, bcnt, N);
  k_pool_div<<<nb(Bz * 128, 256), 256, 0, stream>>>(zsum, bcnt, z0, Bz);
  k_linear<<<nb(Bz * 256, 256), 256, 0, stream>>>(z0, d1w, d1b, t1, Bz, 128, 256);
  k_bn<<<nb(256, 256), 256, 0, stream>>>(t1, bn1g, bn1b, Bz, 256, 1);
  k_linear<<<nb(Bz * 128, 256), 256, 0, stream>>>(t1, d2w, d2b, t2, Bz, 256, 128);
  k_bn<<<nb(128, 128), 128, 0, stream>>>(t2, bn2g, bn2b, Bz, 128, 1);
  k_linear<<<nb(Bz * CLS, 256), 256, 0, stream>>>(t2, d3w, d3b, zout, Bz, 128, CLS);
  k_bn<<<nb(CLS, 64), 64, 0, stream>>>(zout, bn3g, bn3b, Bz, CLS, 0);
  k_argmax<<<dim3(1), 64, 0, stream>>>(zout, zout + Bz * CLS, Bz, CLS);
}